// ProbabilityScore_78211354460173
// MI455X (gfx1250) — compile-verified
//
#include <hip/hip_runtime.h>
#include <math.h>

#define THREADS 256
#define BLOCKS  2048
#define WAVES   (THREADS / 32)

typedef __attribute__((ext_vector_type(2))) float v2f;
typedef __attribute__((ext_vector_type(4))) float f4;
typedef __attribute__((ext_vector_type(8))) float v8f;

// ---------- helpers ----------

// Order-preserving float -> uint mapping so unsigned atomicMax implements float max.
__device__ __forceinline__ unsigned map_f32(float x) {
    unsigned u = __float_as_uint(x);
    return (u & 0x80000000u) ? ~u : (u | 0x80000000u);
}
__device__ __forceinline__ float unmap_f32(unsigned m) {
    unsigned u = (m & 0x80000000u) ? (m ^ 0x80000000u) : ~m;
    return __uint_as_float(u);
}

// Wave32 max reduction (all lanes get the result).
__device__ __forceinline__ float wave_max(float v) {
    #pragma unroll
    for (int off = 16; off > 0; off >>= 1)
        v = fmaxf(v, __shfl_xor(v, off, 32));
    return v;
}

// Wave32 SUM on the matrix pipe: A(16x4) holds each lane's acc in column K=0
// (lanes 0-15 -> rows, K={0,1}) and K=2 (lanes 16-31, K={2,3}); B = ones.
// D[m][n] = rowsum(m). Lanes 0-15 see rows 0..7 in c[0..7], lanes 16-31 see
// rows 8..15 -> one xor-16 shuffle combines to the full 32-lane sum.
__device__ __forceinline__ float wave_sum_wmma(float acc) {
    v2f a; a[0] = acc;  a[1] = 0.0f;
    v2f b; b[0] = 1.0f; b[1] = 1.0f;
    v8f c = {};
    c = __builtin_amdgcn_wmma_f32_16x16x4_f32(
            /*neg_a=*/false, a, /*neg_b=*/false, b,
            /*c_mod=*/(short)0, c, /*reuse_a=*/false, /*reuse_b=*/false);
    float half = ((c[0] + c[1]) + (c[2] + c[3])) + ((c[4] + c[5]) + (c[6] + c[7]));
    return half + __shfl_xor(half, 16, 32);
}

// b_i = softplus(exp((s-m)/tau) - lam), numerically stable softplus.
__device__ __forceinline__ float bval(float s, float m, float inv_tau, float lam) {
    float e = __expf((s - m) * inv_tau);
    float x = e - lam;
    return fmaxf(x, 0.0f) + log1pf(__expf(-fabsf(x)));
}

// ---------- kernels ----------

__global__ void ps_init_kernel(unsigned* wsu) {
    wsu[0] = 0u;  // below map_f32 of any finite/-inf float
}

__global__ void ps_max_kernel(const float* __restrict__ x, int n, unsigned* __restrict__ wsu) {
    const f4* xp = (const f4*)x;
    int n4 = n >> 2;
    int tid = blockIdx.x * blockDim.x + threadIdx.x;
    int stride = gridDim.x * blockDim.x;
    float m = -3.402823466e38f;
    for (int i = tid; i < n4; i += stride) {
        __builtin_prefetch((const void*)(xp + i + stride), 0, 1);
        f4 v = xp[i];
        m = fmaxf(m, fmaxf(fmaxf(v.x, v.y), fmaxf(v.z, v.w)));
    }
    if (blockIdx.x == 0 && threadIdx.x == 0) {      // scalar tail (n % 4)
        for (int i = n4 << 2; i < n; ++i) m = fmaxf(m, x[i]);
    }
    m = wave_max(m);
    __shared__ float smem[WAVES];
    int lane = threadIdx.x & 31, wid = threadIdx.x >> 5;
    if (lane == 0) smem[wid] = m;
    __syncthreads();
    if (threadIdx.x == 0) {
        float bm = smem[0];
        #pragma unroll
        for (int w = 1; w < WAVES; ++w) bm = fmaxf(bm, smem[w]);
        atomicMax(wsu, map_f32(bm));
    }
}

__global__ void ps_sum_kernel(const float* __restrict__ x, int n,
                              const float* __restrict__ tau, const float* __restrict__ lam,
                              const unsigned* __restrict__ wsu,
                              float* __restrict__ partials) {
    float m = unmap_f32(wsu[0]);
    float inv_tau = 1.0f / tau[0];
    float l = lam[0];
    const f4* xp = (const f4*)x;
    int n4 = n >> 2;
    int tid = blockIdx.x * blockDim.x + threadIdx.x;
    int stride = gridDim.x * blockDim.x;
    float acc = 0.0f;
    for (int i = tid; i < n4; i += stride) {
        f4 v = xp[i];
        acc += ((bval(v.x, m, inv_tau, l) + bval(v.y, m, inv_tau, l)) +
                (bval(v.z, m, inv_tau, l) + bval(v.w, m, inv_tau, l)));
    }
    if (blockIdx.x == 0 && threadIdx.x == 0) {      // scalar tail
        for (int i = n4 << 2; i < n; ++i) acc += bval(x[i], m, inv_tau, l);
    }
    float wsum = wave_sum_wmma(acc);                // all 32 lanes active here
    __shared__ float smem[WAVES];
    int lane = threadIdx.x & 31, wid = threadIdx.x >> 5;
    if (lane == 0) smem[wid] = wsum;
    __syncthreads();
    if (threadIdx.x == 0) {
        float s = 0.0f;
        #pragma unroll
        for (int w = 0; w < WAVES; ++w) s += smem[w];
        partials[blockIdx.x] = s;                   // deterministic: fixed tree
    }
}

__global__ void ps_reduce_kernel(const float* __restrict__ partials, int nblk,
                                 float* __restrict__ sum_out) {
    float acc = 0.0f;
    for (int i = threadIdx.x; i < nblk; i += THREADS) acc += partials[i];
    float wsum = wave_sum_wmma(acc);
    __shared__ float smem[WAVES];
    int lane = threadIdx.x & 31, wid = threadIdx.x >> 5;
    if (lane == 0) smem[wid] = wsum;
    __syncthreads();
    if (threadIdx.x == 0) {
        float s = 0.0f;
        #pragma unroll
        for (int w = 0; w < WAVES; ++w) s += smem[w];
        sum_out[0] = s;
    }
}

__global__ void ps_norm_kernel(const float* __restrict__ x, int n,
                               const float* __restrict__ tau, const float* __restrict__ lam,
                               const unsigned* __restrict__ wsu,
                               const float* __restrict__ sum_in,
                               float* __restrict__ out) {
    float m = unmap_f32(wsu[0]);
    float inv_tau = 1.0f / tau[0];
    float l = lam[0];
    float inv_s = 1.0f / sum_in[0];
    const f4* xp = (const f4*)x;
    f4* op = (f4*)out;
    int n4 = n >> 2;
    int tid = blockIdx.x * blockDim.x + threadIdx.x;
    int stride = gridDim.x * blockDim.x;
    for (int i = tid; i < n4; i += stride) {
        f4 v = __builtin_nontemporal_load(xp + i);  // last use of scores
        f4 r;
        r.x = bval(v.x, m, inv_tau, l) * inv_s;
        r.y = bval(v.y, m, inv_tau, l) * inv_s;
        r.z = bval(v.z, m, inv_tau, l) * inv_s;
        r.w = bval(v.w, m, inv_tau, l) * inv_s;
        __builtin_nontemporal_store(r, op + i);     // don't evict L2-resident scores
    }
    if (blockIdx.x == 0 && threadIdx.x == 0) {      // scalar tail
        for (int i = n4 << 2; i < n; ++i)
            out[i] = bval(x[i], m, inv_tau, l) * inv_s;
    }
}

// ---------- launcher ----------

extern "C" void kernel_launch(void* const* d_in, const int* in_sizes, int n_in,
                              void* d_out, int out_size, void* d_ws, size_t ws_size,
                              hipStream_t stream) {
    const float* scores = (const float*)d_in[0];
    const float* tau    = (const float*)d_in[1];
    const float* lam    = (const float*)d_in[2];
    float* out = (float*)d_out;
    int n = in_sizes[0];

    unsigned* wsu    = (unsigned*)d_ws;   // [0]: mapped max
    float* wsf       = (float*)d_ws;      // [1]: final sum
    float* sum_out   = wsf + 1;
    float* partials  = wsf + 16;          // [16 .. 16+BLOCKS): per-block partial sums

    ps_init_kernel<<<1, 1, 0, stream>>>(wsu);
    ps_max_kernel<<<BLOCKS, THREADS, 0, stream>>>(scores, n, wsu);
    ps_sum_kernel<<<BLOCKS, THREADS, 0, stream>>>(scores, n, tau, lam, wsu, partials);
    ps_reduce_kernel<<<1, THREADS, 0, stream>>>(partials, BLOCKS, sum_out);
    ps_norm_kernel<<<BLOCKS, THREADS, 0, stream>>>(scores, n, tau, lam, wsu, sum_out, out);
}